// SOMLayer_15109694948069
// MI455X (gfx1250) — compile-verified
//
#include <hip/hip_runtime.h>
#include <math.h>

typedef __attribute__((ext_vector_type(2))) float v2f;
typedef __attribute__((ext_vector_type(8))) float v8f;

#define BB 16
#define TT 2048
#define DD 256
#define NN 256
#define BT (BB * TT)          // 32768
#define DECAY 0.9f

// ---------- output layout (floats) ----------
// som_z : [0, 8388608)
// scalars (total, kl, div, ts, nb) : [8388608, 8388613)
// q     : [8388613, 16777221)
// bmu   : [16777221, 16809989)

// ---------- workspace layout (bytes) ----------
// colsum f32[256]    @ 0
// znorm2 f32[32768]  @ 1024
// nnorm2 f32[256]    @ 132096
// bmu_i  i32[32768]  @ 133120
// accum  f64[4]      @ 264192   (kl, div, ts, nb)

__device__ __forceinline__ float blockReduceSum256(float v, float* sd) {
    int tid = threadIdx.x;
    sd[tid] = v;
    __syncthreads();
    for (int s = 128; s > 0; s >>= 1) {
        if (tid < s) sd[tid] += sd[tid + s];
        __syncthreads();
    }
    float r = sd[0];
    __syncthreads();
    return r;
}

__global__ void k_zero(float* colsum, double* accum) {
    int t = threadIdx.x;
    if (t < 256) colsum[t] = 0.f;
    if (t < 4) accum[t] = 0.0;
}

__global__ void k_rownorm(const float* __restrict__ x, float* __restrict__ out) {
    __shared__ float sd[256];
    int r = blockIdx.x;
    float v = x[(size_t)r * DD + threadIdx.x];
    float s = blockReduceSum256(v * v, sd);
    if (threadIdx.x == 0) out[r] = s;
}

// Main kernel: 1 wave per 16-row tile. WMMA fp32 GEMM z @ nodes^T, epilogue:
// dist, q (L1-normalized), BMU argmin, q^2 column-sum accumulation.
__global__ __launch_bounds__(128) void k_dist(
        const float* __restrict__ z, const float* __restrict__ nodes,
        const float* __restrict__ znorm2, const float* __restrict__ nnorm2,
        float* __restrict__ qout, float* __restrict__ bmu_f,
        int* __restrict__ bmu_i, float* __restrict__ colsum) {
    __shared__ float s_col[NN];
    int tid = threadIdx.x;
    for (int t = tid; t < NN; t += 128) s_col[t] = 0.f;
    __syncthreads();

    int wave = blockIdx.x * 4 + (tid >> 5);
    int lane = tid & 31;
    int R = wave * 16;
    int n0 = lane & 15;
    int hl = lane >> 4;        // which 16-lane half
    int koff = hl * 2;         // K sub-offset per ISA 16x4 fp32 A layout

    const float* zrow = z + (size_t)(R + n0) * DD;

    v8f acc[16];
#pragma unroll
    for (int j = 0; j < 16; ++j) {
#pragma unroll
        for (int e = 0; e < 8; ++e) acc[j][e] = 0.f;
    }

    for (int k0 = 0; k0 < DD; k0 += 4) {
        v2f a;
        a.x = zrow[k0 + koff];
        a.y = zrow[k0 + koff + 1];
#pragma unroll
        for (int j = 0; j < 16; ++j) {
            const float* nrow = nodes + (size_t)(j * 16 + n0) * DD + k0 + koff;
            v2f b;
            b.x = nrow[0];
            b.y = nrow[1];
            acc[j] = __builtin_amdgcn_wmma_f32_16x16x4_f32(
                false, a, false, b, (short)0, acc[j], false, false);
        }
    }

    float cs[16];
#pragma unroll
    for (int j = 0; j < 16; ++j) cs[j] = 0.f;

#pragma unroll
    for (int v = 0; v < 8; ++v) {
        int r = R + v + hl * 8;          // global row of this D-VGPR
        int t = r & (TT - 1);
        float tw = powf(DECAY, (float)(TT - 1 - t));
        float zn2 = znorm2[r] * tw * tw;
        float rsum = 0.f;
        float best = 3.4e38f;
        int bidx = 0;
        float qv[16];
#pragma unroll
        for (int j = 0; j < 16; ++j) {
            int col = j * 16 + n0;
            float d2 = zn2 + nnorm2[col] - 2.f * tw * acc[j][v];
            d2 = fmaxf(d2, 0.f);
            float dist = (d2 > 0.f) ? sqrtf(d2) : 0.f;
            float qq = 1.f / (1.f + dist);   // ALPHA=1 -> exponent 1
            qv[j] = qq;
            rsum += qq;
            if (dist < best) { best = dist; bidx = col; }
        }
        // reduce across the 16 lanes of this half (rows live in one half)
#pragma unroll
        for (int mask = 1; mask <= 8; mask <<= 1) {
            rsum += __shfl_xor(rsum, mask, 32);
            float ob = __shfl_xor(best, mask, 32);
            int oi = __shfl_xor(bidx, mask, 32);
            if (ob < best || (ob == best && oi < bidx)) { best = ob; bidx = oi; }
        }
        float inv = 1.f / rsum;
        float* qrow = qout + (size_t)r * NN;
#pragma unroll
        for (int j = 0; j < 16; ++j) {
            float qn = qv[j] * inv;
            qrow[j * 16 + n0] = qn;
            cs[j] += qn * qn;
        }
        if (n0 == 0) {
            bmu_f[r] = (float)bidx;
            bmu_i[r] = bidx;
        }
    }
#pragma unroll
    for (int j = 0; j < 16; ++j) atomicAdd(&s_col[j * 16 + n0], cs[j]);
    __syncthreads();
    for (int t = tid; t < NN; t += 128) atomicAdd(&colsum[t], s_col[t]);
}

__global__ void k_div(const float* __restrict__ nodes,
                      const float* __restrict__ nn2, double* accum) {
    __shared__ float sd[256];
    int i = blockIdx.x, j = threadIdx.x;
    const float* a = nodes + (size_t)i * DD;
    const float* b = nodes + (size_t)j * DD;
    float dot = 0.f;
    for (int d = 0; d < DD; ++d) dot += a[d] * b[d];
    float d2 = fmaxf(nn2[i] + nn2[j] - 2.f * dot, 0.f);
    float dist = (d2 > 0.f) ? sqrtf(d2) : 0.f;
    float s = blockReduceSum256(dist, sd);
    if (j == 0) atomicAdd(&accum[1], (double)s);
}

__global__ void k_ts(const float* __restrict__ z, double* accum) {
    __shared__ float sd[256];
    const long long total = (long long)BB * (TT - 1) * DD;
    float local = 0.f;
    for (long long idx = (long long)blockIdx.x * blockDim.x + threadIdx.x;
         idx < total; idx += (long long)gridDim.x * blockDim.x) {
        long long b = idx / ((long long)(TT - 1) * DD);
        long long rem = idx % ((long long)(TT - 1) * DD);
        long long t = rem / DD;
        long long d = rem % DD;
        size_t base = ((size_t)b * TT + t) * DD + d;
        float diff = z[base + DD] - z[base];
        local += diff * diff;
    }
    float s = blockReduceSum256(local, sd);
    if (threadIdx.x == 0) atomicAdd(&accum[2], (double)s);
}

__global__ void k_nb(const int* __restrict__ bmu, double* accum) {
    __shared__ float sd[256];
    const int total = BB * (TT - 1);
    float local = 0.f;
    for (int idx = blockIdx.x * blockDim.x + threadIdx.x; idx < total;
         idx += gridDim.x * blockDim.x) {
        int b = idx / (TT - 1);
        int t = idx % (TT - 1);
        int b0 = bmu[b * TT + t];
        int b1 = bmu[b * TT + t + 1];
        int dr = (b0 >> 4) - (b1 >> 4); dr = dr < 0 ? -dr : dr;
        int dc = (b0 & 15) - (b1 & 15); dc = dc < 0 ? -dc : dc;
        local += (float)(dr + dc);
    }
    float s = blockReduceSum256(local, sd);
    if (threadIdx.x == 0) atomicAdd(&accum[3], (double)s);
}

__global__ void k_kl(const float* __restrict__ q,
                     const float* __restrict__ colsum, double* accum) {
    __shared__ float sd[256];
    int r = blockIdx.x, j = threadIdx.x;
    float qq = q[(size_t)r * NN + j];
    float pun = qq * qq / colsum[j];
    float ps = blockReduceSum256(pun, sd);
    float p = pun / ps;
    float term = p * (logf(p) - logf(qq));
    float s = blockReduceSum256(term, sd);
    if (j == 0) atomicAdd(&accum[0], (double)s);
}

__global__ void k_som(const float* __restrict__ z, const float* __restrict__ nodes,
                      const int* __restrict__ bmu, float* __restrict__ out) {
    size_t total = (size_t)BT * DD;
    for (size_t idx = (size_t)blockIdx.x * blockDim.x + threadIdx.x; idx < total;
         idx += (size_t)gridDim.x * blockDim.x) {
        size_t r = idx >> 8;
        int d = (int)(idx & 255);
        float zv = z[idx];
        float nv = nodes[(size_t)bmu[r] * DD + d];
        out[idx] = zv + 0.1f * (nv - zv);
    }
}

__global__ void k_final(const double* accum, float* out_scalars) {
    double kl = accum[0] / (double)BT;
    double dv = -accum[1] / (double)(NN * NN);
    double ts = accum[2] / ((double)BB * (TT - 1) * DD) * 0.9;
    double nb = accum[3] / ((double)BB * (TT - 1));
    double total = kl + 0.5 * dv + 0.3 * ts + 0.2 * nb;
    out_scalars[0] = (float)total;
    out_scalars[1] = (float)kl;
    out_scalars[2] = (float)dv;
    out_scalars[3] = (float)ts;
    out_scalars[4] = (float)nb;
}

extern "C" void kernel_launch(void* const* d_in, const int* in_sizes, int n_in,
                              void* d_out, int out_size, void* d_ws, size_t ws_size,
                              hipStream_t stream) {
    const float* z = (const float*)d_in[0];      // [16,2048,256]
    const float* nodes = (const float*)d_in[1];  // [16,16,256] -> [256,256]
    float* out = (float*)d_out;

    float* som     = out;                 // 8388608
    float* scalars = out + 8388608;       // 5
    float* qout    = out + 8388613;       // 8388608
    float* bmu_f   = out + 16777221;      // 32768

    char* ws = (char*)d_ws;
    float* colsum  = (float*)(ws + 0);        // 256 f32
    float* znorm2  = (float*)(ws + 1024);     // 32768 f32
    float* nnorm2  = (float*)(ws + 132096);   // 256 f32
    int*   bmu_i   = (int*)(ws + 133120);     // 32768 i32
    double* accum  = (double*)(ws + 264192);  // 4 f64

    k_zero<<<1, 256, 0, stream>>>(colsum, accum);
    k_rownorm<<<BT, 256, 0, stream>>>(z, znorm2);
    k_rownorm<<<NN, 256, 0, stream>>>(nodes, nnorm2);
    k_dist<<<BT / 16 / 4, 128, 0, stream>>>(z, nodes, znorm2, nnorm2,
                                            qout, bmu_f, bmu_i, colsum);
    k_div<<<NN, 256, 0, stream>>>(nodes, nnorm2, accum);
    k_ts<<<2048, 256, 0, stream>>>(z, accum);
    k_nb<<<128, 256, 0, stream>>>(bmu_i, accum);
    k_kl<<<BT, 256, 0, stream>>>(qout, colsum, accum);
    k_som<<<8192, 256, 0, stream>>>(z, nodes, bmu_i, som);
    k_final<<<1, 1, 0, stream>>>(accum, scalars);
}